// PrototypeWiseContrastiveLoss_36807869727391
// MI455X (gfx1250) — compile-verified
//
#include <hip/hip_runtime.h>
#include <math.h>

// ---------------------------------------------------------------------------
// PrototypeWiseContrastiveLoss for MI455X (gfx1250, wave32)
//
// features  : [1, 256, 768, 768] fp32  (604 MB -> HBM-bound, read ONCE)
// prototypes: [21, 256] fp32
// outputs   : unused by reference
// labels    : [1, 1, 768, 768] int32
// out       : scalar fp32 loss
//
// Roofline: 604 MB single pass / 23.3 TB/s ~= 26 us floor; all arithmetic
// (1 fma + 1 mul + 1 ds_add per element) is far below VALU/LDS ceilings.
// Main kernel therefore double-buffers GLOBAL_LOAD_ASYNC_TO_LDS_B128 DMA
// against LDS compute so HBM streams continuously.
// ---------------------------------------------------------------------------

#define FEAT_DIM    256
#define NUM_CLASSES 21
#define IMG_HW      (768 * 768)            // 589824 pixels
#define CHUNK_P     32                     // pixels per LDS tile
#define NCHUNK      (IMG_HW / CHUNK_P)     // 18432 chunks
#define MAIN_BLOCKS 1152
#define CHUNKS_PER_BLOCK (NCHUNK / MAIN_BLOCKS)   // 16, exact
#define MAIN_THREADS 256
#define CPAD        32                     // classes padded to 32 for WMMA

typedef __attribute__((ext_vector_type(2))) float v2f;
typedef __attribute__((ext_vector_type(8))) float v8f;

// ---------------------------------------------------------------------------
// Kernel 1: zero the accumulator workspace (class sums + counts)
// ---------------------------------------------------------------------------
__global__ void zero_ws_kernel(float* __restrict__ ws, int n) {
    int i = blockIdx.x * blockDim.x + threadIdx.x;
    if (i < n) ws[i] = 0.0f;
}

// ---------------------------------------------------------------------------
// Kernel 2: fused per-pixel L2-normalize + scatter-sum into class sums.
// Double-buffered async staging: while chunk k is normalized/scattered from
// tile[cur], chunk k+1 streams into tile[cur^1] via
// GLOBAL_LOAD_ASYNC_TO_LDS_B128 (ASYNCcnt). Async loads complete in order,
// so "s_wait_asynccnt 8" (the 8 just-issued may remain outstanding) proves
// the previous buffer's 8 transfers have landed.
// ---------------------------------------------------------------------------
__global__ __launch_bounds__(MAIN_THREADS)
void class_mean_kernel(const float* __restrict__ features,
                       const int*   __restrict__ labels,
                       float*       __restrict__ gsums,    // [21*256]
                       float*       __restrict__ gcounts)  // [21]
{
    __shared__ float tile[2][FEAT_DIM * CHUNK_P];   // 2 x 32 KB, [d][p]
    __shared__ float lsum[NUM_CLASSES * FEAT_DIM];  // [c][d]  21 KB
    __shared__ float lcount[NUM_CLASSES];
    __shared__ float partial[MAIN_THREADS];
    __shared__ float invn[CHUNK_P];
    __shared__ int   slab[2][CHUNK_P];

    const int t    = threadIdx.x;
    const int p    = t & (CHUNK_P - 1);   // pixel within chunk (0..31)
    const int dblk = t >> 5;              // dim block (0..7), 32 dims each
    const int p4   = (t & 7) * 4;         // 4-float group within a 32-px row

    // 32-bit LDS offsets (generic LDS addr low 32 bits == DS offset)
    const unsigned lds_t0 = (unsigned)(uintptr_t)&tile[0][0];
    const unsigned lds_t1 = (unsigned)(uintptr_t)&tile[1][0];

    for (int i = t; i < NUM_CLASSES * FEAT_DIM; i += MAIN_THREADS) lsum[i] = 0.0f;
    if (t < NUM_CLASSES) lcount[t] = 0.0f;

    // ---- issue helper: 8 x B128 per lane = one 32 KB chunk tile ----
    auto issue_chunk = [&](int buf, int chunk) {
        const int base = chunk * CHUNK_P;
        const unsigned lbase = buf ? lds_t1 : lds_t0;
        #pragma unroll
        for (int i = 0; i < 8; ++i) {
            const int flt = i * 1024 + t * 4;      // flat float idx = d*32 + p4
            const int d   = flt >> 5;              // feature dim (0..255)
            const float* gsrc = features + (size_t)d * IMG_HW + base + p4;
            const unsigned ldsa = lbase + (unsigned)flt * 4u;
            asm volatile("global_load_async_to_lds_b128 %0, %1, off"
                         :: "v"(ldsa),
                            "v"((unsigned long long)(uintptr_t)gsrc)
                         : "memory");
        }
        if (t < CHUNK_P) slab[buf][t] = labels[base + t];
    };

    // prologue: start the pipeline (lsum zeroing above needs no barrier vs.
    // async writes to tile; the first compute barrier below covers both)
    issue_chunk(0, blockIdx.x);

    for (int ci = 0; ci < CHUNKS_PER_BLOCK; ++ci) {
        const int cur = ci & 1;

        // kick off next chunk's DMA into the other buffer, then wait for ours
        if (ci + 1 < CHUNKS_PER_BLOCK) {
            issue_chunk(cur ^ 1, blockIdx.x + (ci + 1) * MAIN_BLOCKS);
            asm volatile("s_wait_asynccnt 0x8" ::: "memory");  // prev 8 done
        } else {
            asm volatile("s_wait_asynccnt 0x0" ::: "memory");
        }
        __syncthreads();

        const float* tcur = tile[cur];

        // --- sum of squares: 8 partials per pixel, conflict-free LDS reads ---
        float s = 0.0f;
        #pragma unroll
        for (int j = 0; j < 32; ++j) {
            float v = tcur[(dblk * 32 + j) * CHUNK_P + p];
            s = fmaf(v, v, s);
        }
        partial[t] = s;
        __syncthreads();

        if (t < CHUNK_P) {
            float tot = 0.0f;
            #pragma unroll
            for (int k = 0; k < 8; ++k) tot += partial[k * CHUNK_P + t];
            invn[t] = 1.0f / fmaxf(sqrtf(tot), 1e-12f);       // F.normalize eps
            atomicAdd(&lcount[slab[cur][t]], 1.0f);            // ds_add_f32
        }
        __syncthreads();

        // --- scatter: lane-rotated d so all 32 lanes hit distinct LDS banks ---
        {
            const int   c   = slab[cur][p];
            const float inv = invn[p];
            #pragma unroll 8
            for (int j = 0; j < 32; ++j) {
                int d = dblk * 32 + ((p + j) & 31);
                atomicAdd(&lsum[c * FEAT_DIM + d], tcur[d * CHUNK_P + p] * inv);
            }
        }
        __syncthreads();   // all reads of tile[cur] done before it is refilled
    }

    // --- flush block-local sums to global accumulators ---
    for (int i = t; i < NUM_CLASSES * FEAT_DIM; i += MAIN_THREADS)
        atomicAdd(&gsums[i], lsum[i]);
    if (t < NUM_CLASSES) atomicAdd(&gcounts[t], lcount[t]);
}

// ---------------------------------------------------------------------------
// Kernel 3: logits = (mean / TEMP) @ prototypes^T via V_WMMA_F32_16X16X4_F32
// (C padded 21->32: 2x2 tiles of 16x16, one wave per tile, K=256 in steps of
// 4), then masked contrastive loss over classes 1..20. B fragments load
// branch-free (clamped column + 0/1 mask folded into v_pk_mul_f32).
// ---------------------------------------------------------------------------
__global__ __launch_bounds__(128)
void finalize_kernel(const float* __restrict__ gsums,
                     const float* __restrict__ gcounts,
                     const float* __restrict__ prototypes,
                     float*       __restrict__ out)
{
    __shared__ float meanA[FEAT_DIM * CPAD];   // [d][c], pre-scaled by 1/TEMP
    __shared__ float logitsS[CPAD * CPAD];
    __shared__ float red[32];

    const int t = threadIdx.x;

    // mean * (1/TEMP), zero-padded for c >= 21; stored d-major for bank spread
    for (int i = t; i < FEAT_DIM * CPAD; i += 128) {
        int d = i >> 5, c = i & 31;
        float v = 0.0f;
        if (c < NUM_CLASSES)
            v = gsums[c * FEAT_DIM + d] * (10.0f / fmaxf(gcounts[c], 1.0f));
        meanA[d * CPAD + c] = v;
    }
    __syncthreads();

    // one wave per 16x16 output tile: wv -> (mt, nt)
    const int lane = t & 31;
    const int wv   = t >> 5;
    const int mt   = wv >> 1, nt = wv & 1;
    const int mr   = lane & 15;        // row/col within tile
    const int kh   = lane >> 4;        // K-half select (lanes 16-31 -> K+2)

    // lane-constant B column: clamp + mask instead of predicated loads
    const int   col  = nt * 16 + mr;
    const int   safe = (col < NUM_CLASSES) ? col : 0;
    const float msk  = (col < NUM_CLASSES) ? 1.0f : 0.0f;
    const v2f*  prow = (const v2f*)(prototypes + safe * FEAT_DIM);

    v8f acc = {};
    for (int k0 = 0; k0 < FEAT_DIM; k0 += 4) {
        const int k = k0 + 2 * kh;
        // A fragment: 16x4 f32, M on lanes, K split across VGPRs/lane-halves
        v2f a;
        a.x = meanA[k * CPAD + (mt * 16 + mr)];
        a.y = meanA[(k + 1) * CPAD + (mt * 16 + mr)];
        // B fragment: 4x16 f32, B[k][n] = prototypes[n][k]; one b64 load
        v2f b = prow[(k0 >> 1) + kh];
        b.x *= msk;
        b.y *= msk;
        acc = __builtin_amdgcn_wmma_f32_16x16x4_f32(
            /*neg_a=*/false, a, /*neg_b=*/false, b,
            /*c_mod=*/(short)0, acc, /*reuse_a=*/false, /*reuse_b=*/false);
    }

    // C/D layout: VGPR v, lanes 0-15 -> M=v, lanes 16-31 -> M=v+8
    #pragma unroll
    for (int v = 0; v < 8; ++v) {
        int row = mt * 16 + v + 8 * kh;
        int c2  = nt * 16 + mr;
        logitsS[row * CPAD + c2] = acc[v];
    }
    __syncthreads();

    // masked contrastive loss over rows/cols 1..20 (row-max over full 0..20)
    if (t < NUM_CLASSES - 1) {
        const int i = t + 1;
        float mx = -1e30f;
        for (int j = 0; j < NUM_CLASSES; ++j)
            mx = fmaxf(mx, logitsS[i * CPAD + j]);
        float denom = 0.0f;
        for (int j = 1; j < NUM_CLASSES; ++j)
            denom += expf(logitsS[i * CPAD + j] - mx);
        red[t] = logf(denom) - (logitsS[i * CPAD + i] - mx);
    }
    __syncthreads();
    if (t == 0) {
        float s = 0.0f;
        for (int k = 0; k < NUM_CLASSES - 1; ++k) s += red[k];
        out[0] = s / (float)(NUM_CLASSES - 1);
    }
}

// ---------------------------------------------------------------------------
extern "C" void kernel_launch(void* const* d_in, const int* in_sizes, int n_in,
                              void* d_out, int out_size, void* d_ws, size_t ws_size,
                              hipStream_t stream) {
    const float* features   = (const float*)d_in[0];   // [1,256,768,768]
    const float* prototypes = (const float*)d_in[1];   // [21,256]
    // d_in[2] (outputs) is unused by the reference computation
    const int*   labels     = (const int*)d_in[3];     // [1,1,768,768]

    float* ws      = (float*)d_ws;
    float* gsums   = ws;                                // 21*256 floats
    float* gcounts = ws + NUM_CLASSES * FEAT_DIM;       // 21 floats

    const int nz = NUM_CLASSES * FEAT_DIM + NUM_CLASSES;
    zero_ws_kernel<<<(nz + 255) / 256, 256, 0, stream>>>(ws, nz);

    class_mean_kernel<<<MAIN_BLOCKS, MAIN_THREADS, 0, stream>>>(
        features, labels, gsums, gcounts);

    finalize_kernel<<<1, 128, 0, stream>>>(
        gsums, gcounts, prototypes, (float*)d_out);
}